// ProbAttention_197568496200
// MI455X (gfx1250) — compile-verified
//
#include <hip/hip_runtime.h>

typedef __attribute__((ext_vector_type(2))) float v2f;
typedef __attribute__((ext_vector_type(8))) float v8f;

// One workgroup (1 wave of 32 lanes) per head. L=128, H=8, u<=32.
// All reshapes in the reference are flat reinterpretations: head h owns the
// contiguous slice [h*128, h*128+128) of q/k/v AND of the (B,H,S,E) output.
__global__ __launch_bounds__(32) void prob_attn_kernel(
    const float* __restrict__ Q, const float* __restrict__ K,
    const float* __restrict__ V, const int* __restrict__ kidx,
    float* __restrict__ out, int u)
{
    constexpr int L = 128;
    const int h   = blockIdx.x;
    const int tid = threadIdx.x;

    const float* q = Q + h * L;
    const float* k = K + h * L;
    const float* v = V + h * L;
    float*       o = out + h * L;

    __shared__ float ldsKs[32];
    __shared__ float ldsM[L];

    // ---- Stage 1: gather Ks = k[kseq_idx] (cols >= u stay 0, masked later) ----
    float ksv = 0.0f;
    if (tid < u) ksv = k[kidx[tid]];
    ldsKs[tid] = ksv;
    __syncthreads();

    const int  col     = tid & 15;
    const bool lowHalf = tid < 16;
    const float NEG_INF = -__builtin_inff();

    // ---- Stage 2: M[s] = max_j(Ks[j]*q[s]) - (sum_j Ks[j]*q[s]) / 128 ----
    // Transposed rank-1 product via V_WMMA_F32_16X16X4_F32:
    //   A (16x4) = Ks rows (j), K=0 column only; two j-tiles (j 0..15, 16..31).
    //   B (4x16) = q columns (s), K=0 row only; eight s-tiles cover s 0..127.
    // D'[j,s]: VGPR m holds row j = jtile*16 + m (+8 for lanes 16..31),
    // column s = stile*16 + (lane & 15).  => per-column reduction is an
    // in-lane tree over 8 VGPRs + one shfl_xor(16) to merge the two halves.
    v2f a0 = {0.0f, 0.0f};
    v2f a1 = {0.0f, 0.0f};
    if (lowHalf) { a0.x = ldsKs[col]; a1.x = ldsKs[16 + col]; }

    const int joff = lowHalf ? 0 : 8;   // row offset of this half inside a tile

    // Preload this lane's q values for the 8 B tiles (coalesced, registers only).
    float qv[8];
    #pragma unroll
    for (int t = 0; t < 8; ++t) qv[t] = q[t * 16 + col];

    #pragma unroll
    for (int t = 0; t < 8; ++t) {
        v2f b = {0.0f, 0.0f};
        if (lowHalf) b.x = qv[t];

        v8f c0 = {};  // rows j = 0..15
        v8f c1 = {};  // rows j = 16..31 (only j < u valid)
        c0 = __builtin_amdgcn_wmma_f32_16x16x4_f32(
                 false, a0, false, b, (short)0, c0, false, false);
        c1 = __builtin_amdgcn_wmma_f32_16x16x4_f32(
                 false, a1, false, b, (short)0, c1, false, false);

        float vmax = NEG_INF;
        float vsum = 0.0f;
        #pragma unroll
        for (int m = 0; m < 8; ++m) {
            const bool v0 = (m + joff) < u;        // j of c0[m] in this half
            const bool v1 = (16 + m + joff) < u;   // j of c1[m] in this half
            const float x0 = c0[m], x1 = c1[m];
            vmax = fmaxf(vmax, v0 ? x0 : NEG_INF);
            vmax = fmaxf(vmax, v1 ? x1 : NEG_INF);
            vsum += v0 ? x0 : 0.0f;
            vsum += v1 ? x1 : 0.0f;
        }
        // merge the two 8-row halves of each column (lane l <-> lane l+16)
        vmax = fmaxf(vmax, __shfl_xor(vmax, 16, 32));
        vsum += __shfl_xor(vsum, 16, 32);

        if (lowHalf) ldsM[t * 16 + col] = vmax - vsum * (1.0f / 128.0f);
    }
    __syncthreads();

    // ---- Stage 3: prefix sums of v (wave32 shuffle scan) ----
    const int s0 = tid * 4;
    float p[4];
    float run = 0.0f;
    #pragma unroll
    for (int i = 0; i < 4; ++i) { run += v[s0 + i]; p[i] = run; }
    const float laneSum = run;
    float x = laneSum;
    #pragma unroll
    for (int off = 1; off < 32; off <<= 1) {
        float y = __shfl_up(x, off, 32);
        if (tid >= off) x += y;
    }
    const float excl = x - laneSum;   // exclusive prefix over earlier lanes

    // ---- Stage 4: top-u membership by rank (jax.lax.top_k tie-break) ----
    float Ms[4];
    int   cnt[4] = {0, 0, 0, 0};
    #pragma unroll
    for (int i = 0; i < 4; ++i) Ms[i] = ldsM[s0 + i];
    for (int t = 0; t < L; ++t) {
        const float Mt = ldsM[t];   // one LDS load, 4 comparisons
        #pragma unroll
        for (int i = 0; i < 4; ++i)
            cnt[i] += (Mt > Ms[i]) || ((Mt == Ms[i]) && (t < s0 + i));
    }

    #pragma unroll
    for (int i = 0; i < 4; ++i) {
        const int s = s0 + i;
        const bool sel = cnt[i] < u;              // s is in the top-u set
        const float pref = excl + p[i];           // inclusive prefix sum of v
        o[s] = sel ? (pref / (float)(s + 1))      // uniform causal mean (softmax of {0,-inf})
                   : v[s];                        // cumsum over size-1 E axis == identity
    }
}

extern "C" void kernel_launch(void* const* d_in, const int* in_sizes, int n_in,
                              void* d_out, int out_size, void* d_ws, size_t ws_size,
                              hipStream_t stream) {
    const float* Q    = (const float*)d_in[0];
    const float* K    = (const float*)d_in[1];
    const float* V    = (const float*)d_in[2];
    const int*   kidx = (const int*)d_in[3];
    float*       out  = (float*)d_out;

    const int u = in_sizes[3];   // = FACTOR * ceil(log(L)) = 25
    const int H = 8;

    prob_attn_kernel<<<H, 32, 0, stream>>>(Q, K, V, kidx, out, u);
}